// MultiScaleHeightDistributionAnalyzer_27728308863852
// MI455X (gfx1250) — compile-verified
//
#include <hip/hip_runtime.h>

#define NB 50
#define BATCH 32

typedef __attribute__((ext_vector_type(2))) float v2f;
typedef __attribute__((ext_vector_type(8))) float v8f;

__device__ __forceinline__ int bin_idx(float v) {
    int i = (int)(v * 0.5f);           // NB/MAXH = 0.5, trunc-toward-zero like jnp astype(int32)
    return i < 0 ? 0 : (i > NB - 1 ? NB - 1 : i);
}

__device__ __forceinline__ float bilinear_sample(const float* __restrict__ Tb,
                                                 int p, int Wo, float sy, float sx) {
    int y = p / Wo, x = p - y * Wo;
    float ry = y * sy, rx = x * sx;
    int y0 = (int)ry, x0 = (int)rx;
    float fy = ry - (float)y0, fx = rx - (float)x0;
    int y1 = min(y0 + 1, 511), x1 = min(x0 + 1, 511);
    const float* r0 = Tb + (size_t)y0 * 512;
    const float* r1 = Tb + (size_t)y1 * 512;
    float top = r0[x0] * (1.f - fx) + r0[x1] * fx;
    float bot = r1[x0] * (1.f - fx) + r1[x1] * fx;
    return top * (1.f - fy) + bot * fy;
}

__global__ void zero_ws_kernel(float* __restrict__ ws, int n) {
    int i = blockIdx.x * blockDim.x + threadIdx.x;
    if (i < n) ws[i] = 0.f;
}

// ---------- scale 0 (512x512, identity resize): vectorized streaming ----------
__global__ void hist_identity_kernel(const float* __restrict__ T, float* __restrict__ histG,
                                     int scale) {
    __shared__ unsigned lh[NB];
    const int b = blockIdx.y;
    const int tid = threadIdx.x;
    if (tid < NB) lh[tid] = 0u;
    __syncthreads();
    const float4* T4 = (const float4*)(T + (size_t)b * 512 * 512);  // 65536 float4 / sample
    const int base = blockIdx.x * 1024;                              // 1024 float4 per block
    #pragma unroll
    for (int k = 0; k < 4; ++k) {
        float4 v = T4[base + k * 256 + tid];                         // global_load_b128
        atomicAdd(&lh[bin_idx(v.x)], 1u);
        atomicAdd(&lh[bin_idx(v.y)], 1u);
        atomicAdd(&lh[bin_idx(v.z)], 1u);
        atomicAdd(&lh[bin_idx(v.w)], 1u);
    }
    __syncthreads();
    if (tid < NB) {
        unsigned c = lh[tid];
        if (c) atomicAdd(&histG[(size_t)(scale * BATCH + b) * NB + tid], (float)c);
    }
}

__global__ void gather_identity_kernel(const float* __restrict__ T,
                                       const float* __restrict__ fwG,
                                       float* __restrict__ out, int scale) {
    __shared__ float lw[NB];
    const int b = blockIdx.y;
    const int tid = threadIdx.x;
    if (tid < NB) lw[tid] = fwG[(size_t)(scale * BATCH + b) * NB + tid];
    __syncthreads();
    const float4* T4 = (const float4*)(T + (size_t)b * 512 * 512);
    float4* O4 = (float4*)(out + (size_t)b * 512 * 512);
    const int base = blockIdx.x * 1024;
    #pragma unroll
    for (int k = 0; k < 4; ++k) {
        int p = base + k * 256 + tid;
        float4 v = T4[p];
        float4 o;
        o.x = lw[bin_idx(v.x)];
        o.y = lw[bin_idx(v.y)];
        o.z = lw[bin_idx(v.z)];
        o.w = lw[bin_idx(v.w)];
        O4[p] = o;                                                   // global_store_b128
    }
}

// ---------- downscaled scales: bilinear gather ----------
__global__ void hist_kernel(const float* __restrict__ T, float* __restrict__ histG,
                            int Wo, float sc, int scale) {
    __shared__ unsigned lh[NB];
    const int b = blockIdx.y;
    const int tid = threadIdx.x;
    if (tid < NB) lh[tid] = 0u;
    __syncthreads();
    const float* Tb = T + (size_t)b * 512 * 512;
    const int npix = Wo * Wo;
    const int start = blockIdx.x * (blockDim.x * 16) + tid;
    for (int k = 0; k < 16; ++k) {
        int p = start + k * blockDim.x;
        if (p < npix) {
            float v = bilinear_sample(Tb, p, Wo, sc, sc);
            atomicAdd(&lh[bin_idx(v)], 1u);
        }
    }
    __syncthreads();
    if (tid < NB) {
        unsigned c = lh[tid];
        if (c) atomicAdd(&histG[(size_t)(scale * BATCH + b) * NB + tid], (float)c);
    }
}

__global__ void gather_kernel(const float* __restrict__ T, const float* __restrict__ fwG,
                              float* __restrict__ out, int Wo, float sc, int scale) {
    __shared__ float lw[NB];
    const int b = blockIdx.y;
    const int tid = threadIdx.x;
    if (tid < NB) lw[tid] = fwG[(size_t)(scale * BATCH + b) * NB + tid];
    __syncthreads();
    const float* Tb = T + (size_t)b * 512 * 512;
    const int npix = Wo * Wo;
    float* ob = out + (size_t)b * npix;
    const int start = blockIdx.x * (blockDim.x * 16) + tid;
    for (int k = 0; k < 16; ++k) {
        int p = start + k * blockDim.x;
        if (p < npix) {
            float v = bilinear_sample(Tb, p, Wo, sc, sc);
            ob[p] = lw[bin_idx(v)];
        }
    }
}

// 4 blocks (one per scale), 128 threads = 4 wave32. Dens -> smoothing -> MLP via
// V_WMMA_F32_16X16X4_F32 (weights staged in LDS, branch-free fragments) ->
// effective-number weights -> normalized bin weights.
__global__ void stats_mlp_kernel(const float* __restrict__ histG,
                                 float* __restrict__ densG,
                                 float* __restrict__ fwG,
                                 const float* __restrict__ W1g, const float* __restrict__ b1g,
                                 const float* __restrict__ W2g, const float* __restrict__ b2g,
                                 const float* __restrict__ W3g, const float* __restrict__ b3g) {
    __shared__ float sDens[BATCH][NB];   // density, later reused for fw
    __shared__ float sA[BATCH][52];      // smoothed activations, K padded 50->52
    __shared__ float sW1[52][64];        // layer-1 weights (rows 50,51 zero)
    __shared__ float sW2[64][32];        // layer-2 weights
    __shared__ float sH1[BATCH][65];     // layer-1 activations (padded stride)
    __shared__ float sH2[BATCH][33];     // layer-2 activations (padded stride)
    __shared__ float sDyn[BATCH];
    __shared__ float sSum[BATCH];

    const int scale = blockIdx.x;
    const int tid = threadIdx.x;
    const int lane = tid & 31;
    const int wave = tid >> 5;
    const int laneM = lane & 15;         // M/N position within tile
    const int laneH = lane >> 4;         // half-select (0: K0/K1 | 1: K2/K3; rows r / r+8)

    // ---- stage weights + counts into LDS (coalesced) ----
    for (int e = tid; e < 52 * 64; e += 128) {
        int k = e >> 6, n = e & 63;
        sW1[k][n] = (k < NB) ? W1g[(size_t)(scale * NB + k) * 64 + n] : 0.f;
    }
    for (int e = tid; e < 64 * 32; e += 128) {
        int k = e >> 5, n = e & 31;
        sW2[k][n] = W2g[(size_t)(scale * 64 + k) * 32 + n];
    }
    for (int e = tid; e < BATCH * NB; e += 128)
        sDens[e / NB][e % NB] = histG[(size_t)scale * BATCH * NB + e];
    __syncthreads();
    // ---- row sums ----
    if (tid < BATCH) {
        float s = 0.f;
        for (int i = 0; i < NB; ++i) s += sDens[tid][i];
        sSum[tid] = s;
    }
    __syncthreads();
    // ---- density ----
    for (int e = tid; e < BATCH * NB; e += 128) {
        int b = e / NB, i = e % NB;
        float s = sSum[b];
        float d = (s > 1e-8f) ? sDens[b][i] / s : (1.0f / NB);
        densG[(size_t)scale * BATCH * NB + e] = d;
        sDens[b][i] = d;
    }
    __syncthreads();
    // ---- LDS smoothing (zero-padded 'same' conv, window from gaussian_filter1d) ----
    {
        const float w5[5] = {0.8582849f, 0.9458279f, 1.0f, 0.9458279f, 0.8582849f};
        for (int e = tid; e < BATCH * 52; e += 128) {
            int b = e / 52, i = e % 52;
            float v = 0.f;
            if (i < NB) {
                #pragma unroll
                for (int k = 0; k < 5; ++k) {
                    int src = i + k - 2;
                    if (src >= 0 && src < NB) v += sDens[b][src] * w5[k];
                }
            }
            sA[b][i] = v;                // K padding columns 50,51 stay zero
        }
    }
    __syncthreads();

    // ---- layer 1: [32 x 50] @ [50 x 64], f32 WMMA 16x16x4, K padded to 52 ----
    {
        const int m0 = (wave & 1) * 16;
        const int nbase = (wave >> 1) * 16;
        const int mrow = m0 + laneM;
        #pragma unroll
        for (int t = 0; t < 2; ++t) {
            const int n0 = nbase + t * 32;
            const int ncol = n0 + laneM;
            v8f acc = {};
            for (int kb = 0; kb < 52; kb += 4) {
                const int k0 = kb + 2 * laneH;
                v2f a, b;
                a.x = sA[mrow][k0];
                a.y = sA[mrow][k0 + 1];
                b.x = sW1[k0][ncol];
                b.y = sW1[k0 + 1][ncol];
                acc = __builtin_amdgcn_wmma_f32_16x16x4_f32(false, a, false, b,
                                                            (short)0, acc, false, false);
            }
            const float bias = b1g[scale * 64 + ncol];
            #pragma unroll
            for (int r = 0; r < 8; ++r) {
                int row = m0 + r + (laneH << 3);
                sH1[row][ncol] = fmaxf(acc[r] + bias, 0.f);
            }
        }
    }
    __syncthreads();

    // ---- layer 2: [32 x 64] @ [64 x 32] ----
    {
        const int m0 = (wave & 1) * 16;
        const int n0 = (wave >> 1) * 16;
        const int mrow = m0 + laneM;
        const int ncol = n0 + laneM;
        v8f acc = {};
        for (int kb = 0; kb < 64; kb += 4) {
            const int k0 = kb + 2 * laneH;
            v2f a, b;
            a.x = sH1[mrow][k0];
            a.y = sH1[mrow][k0 + 1];
            b.x = sW2[k0][ncol];
            b.y = sW2[k0 + 1][ncol];
            acc = __builtin_amdgcn_wmma_f32_16x16x4_f32(false, a, false, b,
                                                        (short)0, acc, false, false);
        }
        const float bias = b2g[scale * 32 + ncol];
        #pragma unroll
        for (int r = 0; r < 8; ++r) {
            int row = m0 + r + (laneH << 3);
            sH2[row][ncol] = fmaxf(acc[r] + bias, 0.f);
        }
    }
    __syncthreads();

    // ---- layer 3 + softplus ----
    if (tid < BATCH) {
        float d = b3g[scale];
        for (int i = 0; i < 32; ++i) d += sH2[tid][i] * W3g[scale * 32 + i];
        sDyn[tid] = (d > 20.f) ? d : log1pf(expf(d));
    }
    __syncthreads();

    // ---- effective-number weights * dyn, mean-normalize ----
    {
        const float lnBeta = -1.0005003335835335e-3f;     // ln(0.999)
        for (int e = tid; e < BATCH * NB; e += 128) {
            int b = e / NB, i = e % NB;
            float sm = sA[b][i];
            float eff = (1.f - expf(sm * 10000.f * lnBeta)) * (1.f / 0.00100001f);
            float hc = 2.f * (float)i + 1.f;              // bin center
            float diff = 1.f / (1.f + expf(-(hc - 20.f) * 0.1f));
            sDens[b][i] = (1.f / (eff + 1e-8f)) * diff * sDyn[b];
        }
        __syncthreads();
        if (tid < BATCH) {
            float s = 0.f;
            for (int i = 0; i < NB; ++i) s += sDens[tid][i];
            sSum[tid] = s * (1.f / NB);
        }
        __syncthreads();
        for (int e = tid; e < BATCH * NB; e += 128) {
            int b = e / NB, i = e % NB;
            fwG[(size_t)scale * BATCH * NB + e] = sDens[b][i] / (sSum[b] + 1e-8f);
        }
    }
}

__global__ void closs_kernel(const float* __restrict__ densG,
                             const float* __restrict__ cw, float* __restrict__ out) {
    __shared__ float red[256];
    const int tid = threadIdx.x;
    const int pi[6] = {0, 0, 0, 1, 1, 2};
    const int pj[6] = {1, 2, 3, 2, 3, 3};
    float acc = 0.f;
    for (int e = tid; e < 6 * BATCH * NB; e += 256) {
        int p = e / (BATCH * NB);
        int r = e - p * (BATCH * NB);
        float di = densG[pi[p] * BATCH * NB + r];
        float dj = densG[pj[p] * BATCH * NB + r];
        if (dj > 0.f)
            acc += dj * (logf(fmaxf(dj, 1e-30f)) - logf(di + 1e-8f));
    }
    red[tid] = acc;
    __syncthreads();
    for (int s = 128; s > 0; s >>= 1) {
        if (tid < s) red[tid] += red[tid + s];
        __syncthreads();
    }
    if (tid == 0) out[0] = red[0] * (1.f / (float)BATCH) * cw[0];
}

extern "C" void kernel_launch(void* const* d_in, const int* in_sizes, int n_in,
                              void* d_out, int out_size, void* d_ws, size_t ws_size,
                              hipStream_t stream) {
    const float* targets = (const float*)d_in[4];
    const float* W1 = (const float*)d_in[5];
    const float* b1 = (const float*)d_in[6];
    const float* W2 = (const float*)d_in[7];
    const float* b2 = (const float*)d_in[8];
    const float* W3 = (const float*)d_in[9];
    const float* b3 = (const float*)d_in[10];
    const float* cw = (const float*)d_in[11];
    float* out = (float*)d_out;
    float* ws = (float*)d_ws;

    float* histG = ws;                       // [4][32][50]
    float* densG = ws + 4 * BATCH * NB;      // [4][32][50]
    float* fwG   = ws + 8 * BATCH * NB;      // [4][32][50]

    zero_ws_kernel<<<(4 * BATCH * NB + 255) / 256, 256, 0, stream>>>(histG, 4 * BATCH * NB);

    // scale 0: identity, vectorized
    {
        dim3 grid(64, BATCH, 1);             // 64 blocks * 1024 float4 = 262144 px
        hist_identity_kernel<<<grid, 256, 0, stream>>>(targets, histG, 0);
    }
    // scales 1..3: bilinear
    const int Hs[3] = {256, 128, 64};
    for (int s = 1; s < 4; ++s) {
        const int Ho = Hs[s - 1];
        const float sc = 511.0f / (float)(Ho - 1);
        dim3 grid((Ho * Ho + 4095) / 4096, BATCH, 1);
        hist_kernel<<<grid, 256, 0, stream>>>(targets, histG, Ho, sc, s);
    }

    stats_mlp_kernel<<<4, 128, 0, stream>>>(histG, densG, fwG, W1, b1, W2, b2, W3, b3);

    closs_kernel<<<1, 256, 0, stream>>>(densG, cw, out + 11141120);

    const size_t off[4] = {0, 8388608, 10485760, 11010048};
    {
        dim3 grid(64, BATCH, 1);
        gather_identity_kernel<<<grid, 256, 0, stream>>>(targets, fwG, out + off[0], 0);
    }
    for (int s = 1; s < 4; ++s) {
        const int Ho = Hs[s - 1];
        const float sc = 511.0f / (float)(Ho - 1);
        dim3 grid((Ho * Ho + 4095) / 4096, BATCH, 1);
        gather_kernel<<<grid, 256, 0, stream>>>(targets, fwG, out + off[s], Ho, sc, s);
    }
}